// CRNN_10943576670318
// MI455X (gfx1250) — compile-verified
//
#include <hip/hip_runtime.h>

// CRNN ODE step, batched B=1048576 (f32 end-to-end; exp() reaches ~1e13 so
// fp16/fp8 WMMA is numerically invalid -> V_WMMA_F32_16X16X4_F32).
//
//   w_v = [log(clip(u)), -1/(R*T), log(T)]            [B,20]
//   inter = clip(w_v @ w_in + w_b, -30, 30)           [B,36]
//   du = clip(exp(inter) @ w_out^T, -1e5, 1e5)        [B,18]
//
// v2 changes vs v1 (from asm inspection):
//  - weights pre-swizzled in LDS into per-lane WMMA fragment layout
//    [kstep][ntile][lane][2] -> every B fragment is ONE aligned ds_load_b64
//    with an immediate offset (kills the v_mov packing + 2addr_b32 pairs)
//  - 2 M-tiles (32 rows) per wave -> each B fragment feeds 2 WMMAs,
//    halving DS traffic + address VALU per row. 66 static WMMAs.
//  - 4 waves / 128 threads per block, 128 rows/block, LDS = 37.3 KB.

typedef __attribute__((ext_vector_type(2))) float v2f;
typedef __attribute__((ext_vector_type(8))) float v8f;

#define NS    18
#define NR    36
#define NSP2  20           // NS + 2
#define R_KCAL 0.0019872036f

#define WAVES   4
#define ROWS_W  32         // rows per wave (2 M tiles)
#define ROWS_B  (WAVES * ROWS_W)   // 128 rows per block

__global__ __launch_bounds__(128) void crnn_wmma_kernel(
    const float* __restrict__ u,      // [B, NS]
    const float* __restrict__ T,      // [B]
    const float* __restrict__ w_in,   // [NSP2, NR] row-major
    const float* __restrict__ w_b,    // [NR]
    const float* __restrict__ w_out,  // [NS, NR] row-major
    float* __restrict__ out)          // [B, NS]
{
    // B fragments of w_in, pre-swizzled: [kk][nt][lane][2]
    //   value(c) = w_in[4*kk + 2*(lane/16) + c][nt*16 + lane%16]   (0 if n>=36)
    __shared__ float s_winfrag[5 * 3 * 32 * 2];    // 960
    __shared__ float s_wb[48];
    // B fragments of w_out^T: [kk][nt][lane][2]
    //   value(c) = w_out[nt*16 + lane%16][4*kk + 2*(lane/16) + c]  (0 if n>=18)
    __shared__ float s_woutfrag[9 * 2 * 32 * 2];   // 1152
    __shared__ float s_wv[WAVES][ROWS_W * NSP2];   // per-wave w_v (reused for out)
    __shared__ float s_e [WAVES][ROWS_W * NR];     // per-wave exp(inter)

    const int tid  = threadIdx.x;
    const int wave = tid >> 5;
    const int lane = tid & 31;
    const int half = lane >> 4;   // 0: lanes 0-15, 1: lanes 16-31
    const int m    = lane & 15;

    // ---- stage 0: weights -> LDS in fragment layout ----
    for (int i = tid; i < 5 * 3 * 32 * 2; i += 128) {
        int c  = i & 1;
        int ln = (i >> 1) & 31;
        int nt = (i >> 6) % 3;
        int kk = (i >> 6) / 3;
        int k  = 4 * kk + 2 * (ln >> 4) + c;          // < 20
        int n  = nt * 16 + (ln & 15);
        s_winfrag[i] = (n < NR) ? w_in[k * NR + n] : 0.0f;
    }
    if (tid < 48) s_wb[tid] = (tid < NR) ? w_b[tid] : 0.0f;
    for (int i = tid; i < 9 * 2 * 32 * 2; i += 128) {
        int c  = i & 1;
        int ln = (i >> 1) & 31;
        int nt = (i >> 6) & 1;
        int kk = (i >> 6) >> 1;
        int k  = 4 * kk + 2 * (ln >> 4) + c;          // < 36
        int n  = nt * 16 + (ln & 15);
        s_woutfrag[i] = (n < NS) ? w_out[n * NR + k] : 0.0f;
    }

    const int rowBase = blockIdx.x * ROWS_B + wave * ROWS_W;

    // ---- stage 1: coalesced u load, clip+log -> w_v tiles (32 rows) ----
    {
        const float* uBase = u + (size_t)rowBase * NS;   // 32*18 = 576 floats
        float* wv = s_wv[wave];
#pragma unroll
        for (int j = 0; j < 18; ++j) {
            int g = lane + 32 * j;                       // 0..575, coalesced
            float x = uBase[g];
            x = fminf(fmaxf(x, 1e-6f), 60.0f);
            int r = g / NS, c = g - r * NS;
            wv[r * NSP2 + c] = __logf(x);
        }
        float t = T[rowBase + lane];                     // lane = row, all 32
        wv[lane * NSP2 + 18] = -1.0f / (R_KCAL * t);
        wv[lane * NSP2 + 19] = __logf(t);
    }
    __syncthreads();

    // ---- stage 2: GEMM1  [32x48] = [32x20] x [20x48], 30 WMMAs ----
    v8f c00 = {}, c01 = {}, c02 = {};   // M tile 0 (rows 0-15)
    v8f c10 = {}, c11 = {}, c12 = {};   // M tile 1 (rows 16-31)
    {
        // A 16x4 f32: lanes 0-15 hold K=2k,2k+1 rowsM ; lanes 16-31 hold K+2
        const float* a0p = &s_wv[wave][ m       * NSP2 + 2 * half];
        const float* a1p = &s_wv[wave][(16 + m) * NSP2 + 2 * half];
        const float* bp  = &s_winfrag[lane * 2];
#pragma unroll
        for (int kk = 0; kk < 5; ++kk) {
            v2f a0 = *(const v2f*)(a0p + 4 * kk);        // imm offset 16*kk
            v2f a1 = *(const v2f*)(a1p + 4 * kk);
            v2f b0 = *(const v2f*)(bp + (kk * 3 + 0) * 64);  // imm 256B stride
            v2f b1 = *(const v2f*)(bp + (kk * 3 + 1) * 64);
            v2f b2 = *(const v2f*)(bp + (kk * 3 + 2) * 64);
            c00 = __builtin_amdgcn_wmma_f32_16x16x4_f32(false, a0, false, b0, (short)0, c00, false, false);
            c01 = __builtin_amdgcn_wmma_f32_16x16x4_f32(false, a0, false, b1, (short)0, c01, false, false);
            c02 = __builtin_amdgcn_wmma_f32_16x16x4_f32(false, a0, false, b2, (short)0, c02, false, false);
            c10 = __builtin_amdgcn_wmma_f32_16x16x4_f32(false, a1, false, b0, (short)0, c10, false, false);
            c11 = __builtin_amdgcn_wmma_f32_16x16x4_f32(false, a1, false, b1, (short)0, c11, false, false);
            c12 = __builtin_amdgcn_wmma_f32_16x16x4_f32(false, a1, false, b2, (short)0, c12, false, false);
        }
    }

    // ---- stage 3: bias + clamp + exp -> E tiles ----
    {
        float* E = s_e[wave];
        const float b0 = s_wb[m], b1 = s_wb[16 + m], b2 = s_wb[32 + m];
#pragma unroll
        for (int v = 0; v < 8; ++v) {
            // C/D layout: VGPR v -> M=v (lanes 0-15) / M=v+8 (lanes 16-31)
            int mr = v + 8 * half;
            float e00 = __expf(fminf(fmaxf(c00[v] + b0, -30.0f), 30.0f));
            float e01 = __expf(fminf(fmaxf(c01[v] + b1, -30.0f), 30.0f));
            float e02 = __expf(fminf(fmaxf(c02[v] + b2, -30.0f), 30.0f));
            float e10 = __expf(fminf(fmaxf(c10[v] + b0, -30.0f), 30.0f));
            float e11 = __expf(fminf(fmaxf(c11[v] + b1, -30.0f), 30.0f));
            float e12 = __expf(fminf(fmaxf(c12[v] + b2, -30.0f), 30.0f));
            E[ mr       * NR + m]      = e00;
            E[ mr       * NR + 16 + m] = e01;
            E[(16 + mr) * NR + m]      = e10;
            E[(16 + mr) * NR + 16 + m] = e11;
            if (m < 4) {                                  // cols 32..35 only
                E[ mr       * NR + 32 + m] = e02;
                E[(16 + mr) * NR + 32 + m] = e12;
            }
        }
    }
    __syncthreads();

    // ---- stage 4: GEMM2  [32x32] = [32x36] x [36x32], 36 WMMAs ----
    v8f d00 = {}, d01 = {};
    v8f d10 = {}, d11 = {};
    {
        const float* a0p = &s_e[wave][ m       * NR + 2 * half];
        const float* a1p = &s_e[wave][(16 + m) * NR + 2 * half];
        const float* bp  = &s_woutfrag[lane * 2];
#pragma unroll
        for (int kk = 0; kk < 9; ++kk) {
            v2f a0 = *(const v2f*)(a0p + 4 * kk);
            v2f a1 = *(const v2f*)(a1p + 4 * kk);
            v2f b0 = *(const v2f*)(bp + (kk * 2 + 0) * 64);
            v2f b1 = *(const v2f*)(bp + (kk * 2 + 1) * 64);
            d00 = __builtin_amdgcn_wmma_f32_16x16x4_f32(false, a0, false, b0, (short)0, d00, false, false);
            d01 = __builtin_amdgcn_wmma_f32_16x16x4_f32(false, a0, false, b1, (short)0, d01, false, false);
            d10 = __builtin_amdgcn_wmma_f32_16x16x4_f32(false, a1, false, b0, (short)0, d10, false, false);
            d11 = __builtin_amdgcn_wmma_f32_16x16x4_f32(false, a1, false, b1, (short)0, d11, false, false);
        }
    }

    // ---- stage 5: clamp, stage through LDS (reuse w_v), coalesced store ----
    {
        float* OB = s_wv[wave];   // needs 32*18 = 576 <= 640 floats
#pragma unroll
        for (int v = 0; v < 8; ++v) {
            int mr = v + 8 * half;
            OB[ mr       * NS + m] = fminf(fmaxf(d00[v], -1e5f), 1e5f);
            OB[(16 + mr) * NS + m] = fminf(fmaxf(d10[v], -1e5f), 1e5f);
            if (m < 2) {
                OB[ mr       * NS + 16 + m] = fminf(fmaxf(d01[v], -1e5f), 1e5f);
                OB[(16 + mr) * NS + 16 + m] = fminf(fmaxf(d11[v], -1e5f), 1e5f);
            }
        }
    }
    __syncthreads();
    {
        const float* OB = s_wv[wave];
        float* oBase = out + (size_t)rowBase * NS;
#pragma unroll
        for (int j = 0; j < 18; ++j) {
            int g = lane + 32 * j;        // coalesced 576-float slab
            oBase[g] = OB[g];
        }
    }
}

extern "C" void kernel_launch(void* const* d_in, const int* in_sizes, int n_in,
                              void* d_out, int out_size, void* d_ws, size_t ws_size,
                              hipStream_t stream) {
    (void)n_in; (void)d_ws; (void)ws_size; (void)out_size;
    const float* u     = (const float*)d_in[0];
    const float* T     = (const float*)d_in[1];
    const float* w_in  = (const float*)d_in[2];
    const float* w_b   = (const float*)d_in[3];
    const float* w_out = (const float*)d_in[4];
    float* out = (float*)d_out;

    const int batch = in_sizes[1];        // T has B elements
    const int grid  = batch / ROWS_B;     // 128 rows per block

    crnn_wmma_kernel<<<grid, 128, 0, stream>>>(u, T, w_in, w_b, w_out, out);
}